// SOGABase_5274219839794
// MI455X (gfx1250) — compile-verified
//
#include <hip/hip_runtime.h>

#define N_NODES 100000
#define N_EDGES 1000000
#define IN_DIM 128
#define HID_DIM 64
#define NUM_CLASSES 10

typedef __attribute__((ext_vector_type(2))) float v2f;
typedef __attribute__((ext_vector_type(8))) float v8f;

// ---------------- degree / normalization ----------------

__global__ void deg_init(float* __restrict__ deg) {
    int i = blockIdx.x * blockDim.x + threadIdx.x;
    if (i < N_NODES) deg[i] = 1.0f;  // self-loop
}

__global__ void deg_count(const int* __restrict__ ei, float* __restrict__ deg) {
    int e = blockIdx.x * blockDim.x + threadIdx.x;
    if (e < N_EDGES) {
        unsafeAtomicAdd(&deg[ei[N_EDGES + e]], 1.0f);  // hardware global_atomic_add_f32
    }
}

__global__ void deg_rsqrt(float* __restrict__ dis) {
    int i = blockIdx.x * blockDim.x + threadIdx.x;
    if (i < N_NODES) dis[i] = rsqrtf(dis[i]);
}

// norm[e] = dis[src]*dis[dst]  (layer-invariant; computed once, reused 3x)
__global__ void edge_norm(const int* __restrict__ ei, const float* __restrict__ dis,
                          float* __restrict__ nrm) {
    int e = blockIdx.x * blockDim.x + threadIdx.x;
    if (e < N_EDGES) {
        nrm[e] = dis[ei[e]] * dis[ei[N_EDGES + e]];
    }
}

// ---------------- fp32 WMMA GEMM with fused self-loop epilogue ----------------
// h[M,NC] = act(A[M,K]) @ W[K,NC]; if SELFLOOP also agg[M,NC] = h*dis[r]^2 + b[c]
// (bias enters exactly once, via agg, per the reference). One wave per block
// computes a 16 x (NTILE*16) strip: A fragment loaded once per k-step, reused
// across NTILE back-to-back v_wmma_f32_16x16x4_f32.
template <int K, int NC, int NTILE, bool RELU, bool GUARD, bool SELFLOOP>
__global__ void gemm_wmma(const float* A,            // may alias agg (in-place safe)
                          const float* __restrict__ W,
                          float* __restrict__ hout,
                          float* agg,
                          const float* __restrict__ dis,
                          const float* __restrict__ bias) {
    const int lane  = threadIdx.x;         // 0..31
    const int mlane = lane & 15;
    const int khalf = lane >> 4;           // 0: K pair {0,1}; 1: K pair {2,3}
    const int row0  = blockIdx.x * 16;
    const int arow  = row0 + mlane;

    v8f acc[NTILE];
#pragma unroll
    for (int nt = 0; nt < NTILE; ++nt) acc[nt] = (v8f){};

    const float* arow_p = A + (size_t)arow * K;

#pragma unroll 4
    for (int k = 0; k < K; k += 4) {
        const int ka = k + 2 * khalf;
        float a0 = arow_p[ka];
        float a1 = arow_p[ka + 1];
        if (RELU) { a0 = fmaxf(a0, 0.0f); a1 = fmaxf(a1, 0.0f); }
        v2f a; a.x = a0; a.y = a1;

#pragma unroll
        for (int nt = 0; nt < NTILE; ++nt) {
            const int bcol = nt * 16 + mlane;
            float b0, b1;
            if (GUARD) {
                // branch-free padding of N to 16: clamp address, zero the lane
                const int   bcs   = bcol < NC ? bcol : NC - 1;
                const float bmask = bcol < NC ? 1.0f : 0.0f;
                b0 = W[(size_t)ka * NC + bcs] * bmask;
                b1 = W[(size_t)(ka + 1) * NC + bcs] * bmask;
            } else {
                b0 = W[(size_t)ka * NC + bcol];
                b1 = W[(size_t)(ka + 1) * NC + bcol];
            }
            v2f b; b.x = b0; b.y = b1;
            acc[nt] = __builtin_amdgcn_wmma_f32_16x16x4_f32(
                false, a, false, b, (short)0, acc[nt], false, false);
        }
    }

    // C/D layout: VGPR v holds row (v + 8*khalf), col = mlane (within tile)
    float d2v[8];
    if (SELFLOOP) {
#pragma unroll
        for (int v = 0; v < 8; ++v) {
            const float dv = dis[row0 + v + 8 * khalf];
            d2v[v] = dv * dv;   // dis^2 = 1/deg
        }
    }
#pragma unroll
    for (int nt = 0; nt < NTILE; ++nt) {
        const int c = nt * 16 + mlane;
        if (GUARD && c >= NC) continue;
        const float badd = bias[c];
#pragma unroll
        for (int v = 0; v < 8; ++v) {
            const int    r   = row0 + v + 8 * khalf;
            const float  val = acc[nt][v];
            if (SELFLOOP) {
                hout[(size_t)r * NC + c] = val;                    // raw messages
                agg [(size_t)r * NC + c] = val * d2v[v] + badd;    // self-loop + bias
            } else {
                hout[(size_t)r * NC + c] = val + badd;             // classifier
            }
        }
    }
}

// ---------------- edge scatter: agg[dst] += h[src] * norm[e] ----------------
// 256 edges staged per block via the CDNA5 async-to-LDS path, then 16 lanes per
// edge gather one 256B row of h (float4 each) and issue hardware f32 atomics.
#define EPB 256
__global__ void edge_scatter(const int* __restrict__ ei, const float* __restrict__ nrm,
                             const float* __restrict__ h, float* __restrict__ agg) {
    __shared__ int   s_src[EPB];
    __shared__ int   s_dst[EPB];
    __shared__ float s_nrm[EPB];

    const int tid = threadIdx.x;
    const int e0  = blockIdx.x * EPB;
    const int e   = e0 + tid;

    if (e < N_EDGES) {
        // ISA: LDS address = low 32 bits of the generic shared pointer
        const unsigned ls = (unsigned)(uintptr_t)&s_src[tid];
        const unsigned ld = (unsigned)(uintptr_t)&s_dst[tid];
        const unsigned ln = (unsigned)(uintptr_t)&s_nrm[tid];
        const int*   gs = ei + e;
        const int*   gd = ei + N_EDGES + e;
        const float* gn = nrm + e;
        asm volatile("global_load_async_to_lds_b32 %0, %1, off" :: "v"(ls), "v"(gs) : "memory");
        asm volatile("global_load_async_to_lds_b32 %0, %1, off" :: "v"(ld), "v"(gd) : "memory");
        asm volatile("global_load_async_to_lds_b32 %0, %1, off" :: "v"(ln), "v"(gn) : "memory");
    }
    asm volatile("s_wait_asynccnt 0x0" ::: "memory");
    __syncthreads();

    const int f     = (tid & 15) << 2;   // feature offset (float4)
    const int esub0 = tid >> 4;          // 0..15

    auto body = [&](int el) {
        const int   s = s_src[el];
        const int   d = s_dst[el];
        const float w = s_nrm[el];
        const float4 hv = *reinterpret_cast<const float4*>(h + (size_t)s * HID_DIM + f);
        float* ap = agg + (size_t)d * HID_DIM + f;
        unsafeAtomicAdd(ap + 0, hv.x * w);
        unsafeAtomicAdd(ap + 1, hv.y * w);
        unsafeAtomicAdd(ap + 2, hv.z * w);
        unsafeAtomicAdd(ap + 3, hv.w * w);
    };

    if (e0 + EPB <= N_EDGES) {
        // full block: guard-free hot path
#pragma unroll 4
        for (int t = 0; t < 16; ++t) body(esub0 + t * 16);
    } else {
        for (int t = 0; t < 16; ++t) {
            const int el = esub0 + t * 16;
            if (e0 + el < N_EDGES) body(el);
        }
    }
}

// ---------------- driver ----------------

extern "C" void kernel_launch(void* const* d_in, const int* in_sizes, int n_in,
                              void* d_out, int out_size, void* d_ws, size_t ws_size,
                              hipStream_t stream) {
    const float* x  = (const float*)d_in[0];
    const int*   ei = (const int*)d_in[1];
    const float* W0 = (const float*)d_in[2];
    const float* b0 = (const float*)d_in[3];
    const float* W1 = (const float*)d_in[4];
    const float* b1 = (const float*)d_in[5];
    const float* W2 = (const float*)d_in[6];
    const float* b2 = (const float*)d_in[7];
    const float* Wc = (const float*)d_in[8];
    const float* bc = (const float*)d_in[9];
    float* out = (float*)d_out;

    float* dis = (float*)d_ws;                          // N_NODES
    float* nrm = dis + N_NODES;                         // N_EDGES
    float* h   = nrm + N_EDGES;                         // N_NODES*64
    float* agg = h + (size_t)N_NODES * HID_DIM;         // N_NODES*64

    const int nthr = 256;
    deg_init <<<(N_NODES + nthr - 1) / nthr, nthr, 0, stream>>>(dis);
    deg_count<<<(N_EDGES + nthr - 1) / nthr, nthr, 0, stream>>>(ei, dis);
    deg_rsqrt<<<(N_NODES + nthr - 1) / nthr, nthr, 0, stream>>>(dis);
    edge_norm<<<(N_EDGES + nthr - 1) / nthr, nthr, 0, stream>>>(ei, dis, nrm);

    const int mtiles  = N_NODES / 16;                   // 6250 (exact)
    const int eblocks = (N_EDGES + EPB - 1) / EPB;      // 3907

    // layer 0: h = x@W0 ; agg = h*dis^2 + b0 (fused) ; scatter
    gemm_wmma<IN_DIM, HID_DIM, 4, false, false, true>
        <<<mtiles, 32, 0, stream>>>(x, W0, h, agg, dis, b0);
    edge_scatter<<<eblocks, EPB, 0, stream>>>(ei, nrm, h, agg);

    // layer 1: h = relu(agg)@W1 (ReLU fused into A-load; in-place agg is safe)
    gemm_wmma<HID_DIM, HID_DIM, 4, true, false, true>
        <<<mtiles, 32, 0, stream>>>(agg, W1, h, agg, dis, b1);
    edge_scatter<<<eblocks, EPB, 0, stream>>>(ei, nrm, h, agg);

    // layer 2
    gemm_wmma<HID_DIM, HID_DIM, 4, true, false, true>
        <<<mtiles, 32, 0, stream>>>(agg, W2, h, agg, dis, b2);
    edge_scatter<<<eblocks, EPB, 0, stream>>>(ei, nrm, h, agg);

    // classifier: out = agg@Wc + bc ; N=10 padded to one 16-wide tile
    gemm_wmma<HID_DIM, NUM_CLASSES, 1, false, true, false>
        <<<mtiles, 32, 0, stream>>>(agg, Wc, out, nullptr, nullptr, bc);
}